// ApproxNDCGLoss_36532991819996
// MI455X (gfx1250) — compile-verified
//
#include <hip/hip_runtime.h>
#include <stdint.h>

// ============================================================================
// ApproxNDCGLoss for N = 16,777,216 (gfx1250 / MI455X)
//
//   approx_dcg = S*T1 + T2        (one fused streaming pass, 3 reductions)
//   idcg       = sum_k g_(k) * log2(k+2)  via bucketed exact ranking:
//     key = (uint)(t * 2^24) << 8   (monotone, injective, ~uniform)
//     65536-bin histogram -> scan -> scatter -> per-bucket O(m^2) tiled rank
//
// Workspace layout (needs 1,335,296 + 4*n bytes ~= 68.4 MB for n = 2^24):
//   [0)        hist      65536 u32
//   [256K)     off       65536 u32   (ascending bucket offsets)
//   [512K)     base      65536 u32   (elements in strictly-greater buckets)
//   [768K)     cursor    65536 u32   (scatter cursors)
//   [1M)       idcg_part 65536 f32
//   [1.25M)    partials  MAIN_WGS*3 f32
//   [1335296)  keys      n u32
// ============================================================================

#define BINS        65536
#define MAIN_WGS    2048
#define MAIN_TPB    256
#define SCAT_WGS    4096
#define SCAT_TPB    256
#define BUCKET_TPB  256
#define TILE        256
#define FINAL_TPB   256

typedef float v2f __attribute__((ext_vector_type(2)));
typedef float v8f __attribute__((ext_vector_type(8)));

__device__ __forceinline__ uint32_t key_of(float t) {
  // targets are float32 on (at least) a 2^-24 grid in [0,1): t*2^24 is exact.
  return ((uint32_t)(t * 16777216.0f)) << 8;
}

// ---- CDNA5 async global->LDS copy (one dword per active lane) ---------------
// Generic LDS pointers carry the LDS byte offset in addr[31:0] (ISA aperture
// rules), so truncation yields the ds address operand for the async op.
__device__ __forceinline__ void async_copy_dword_to_lds(const unsigned int* gsrc,
                                                        unsigned int* lds_dst) {
  unsigned int lds_off = (unsigned int)(uintptr_t)lds_dst;
  unsigned long long ga = (unsigned long long)(uintptr_t)gsrc;
  asm volatile("global_load_async_to_lds_b32 %0, %1, off"
               :
               : "v"(lds_off), "v"(ga)
               : "memory");
}

__device__ __forceinline__ void wait_asynccnt0() {
  asm volatile("s_wait_asynccnt 0" ::: "memory");
}

// ---- wave32 sum via V_WMMA_F32_16X16X4_F32 with B = ones --------------------
// A[m][0] = v(lane m), A[m][2] = v(lane m+16), rest 0  ->  D[m][*] = v_m+v_{m+16}
// Per lane, the 8 C/D VGPRs cover M = 0..7 (lanes<16) or 8..15 (lanes>=16);
// summing them and xor-shuffling across lane 16 yields the full wave sum.
__device__ __forceinline__ float wave_sum32(float v) {
#if __has_builtin(__builtin_amdgcn_wmma_f32_16x16x4_f32)
  v2f a; a[0] = v;    a[1] = 0.0f;
  v2f b; b[0] = 1.0f; b[1] = 1.0f;
  v8f c = {0.0f, 0.0f, 0.0f, 0.0f, 0.0f, 0.0f, 0.0f, 0.0f};
  v8f d = __builtin_amdgcn_wmma_f32_16x16x4_f32(false, a, false, b,
                                                (short)0, c, false, false);
  float s = ((d[0] + d[1]) + (d[2] + d[3])) + ((d[4] + d[5]) + (d[6] + d[7]));
  s += __shfl_xor(s, 16, 32);
  return s;
#else
  for (int o = 16; o > 0; o >>= 1) v += __shfl_xor(v, o, 32);
  return v;
#endif
}

// ---- pass 1: fused streaming reductions + key histogram ---------------------
__global__ void __launch_bounds__(MAIN_TPB)
pass_main(const float* __restrict__ pred, const float* __restrict__ targ,
          unsigned int* __restrict__ hist, float* __restrict__ partials, int n)
{
  const float nm1 = (float)(n - 1);
  const int tid    = threadIdx.x;
  const int gid    = blockIdx.x * MAIN_TPB + tid;
  const int stride = gridDim.x * MAIN_TPB;

  float t1 = 0.0f, t2 = 0.0f, sd = 0.0f;

  const int n4 = n >> 2;
  const float4* p4 = (const float4*)pred;
  const float4* q4 = (const float4*)targ;
  for (int i = gid; i < n4; i += stride) {
    float4 p = p4[i];
    float4 t = q4[i];
    __builtin_prefetch(&p4[i + stride], 0, 0);   // global_prefetch_b8
    __builtin_prefetch(&q4[i + stride], 0, 0);
    const float pv[4] = {p.x, p.y, p.z, p.w};
    const float tv[4] = {t.x, t.y, t.z, t.w};
    const int bidx = i << 2;
#pragma unroll
    for (int u = 0; u < 4; ++u) {
      float e    = expf(pv[u]);
      float inv  = 1.0f / (e + nm1);
      float gain = exp2f(tv[u]) - 1.0f;
      float disc = 1.0f / log2f((float)(bidx + u) + 2.0f);
      t1 += gain * inv;
      t2 += gain * (e - 1.0f) * disc * inv;
      sd += disc;
      atomicAdd(&hist[key_of(tv[u]) >> 16], 1u);
    }
  }
  for (int i = (n4 << 2) + gid; i < n; i += stride) {   // scalar tail
    float e    = expf(pred[i]);
    float tt   = targ[i];
    float inv  = 1.0f / (e + nm1);
    float gain = exp2f(tt) - 1.0f;
    float disc = 1.0f / log2f((float)i + 2.0f);
    t1 += gain * inv;
    t2 += gain * (e - 1.0f) * disc * inv;
    sd += disc;
    atomicAdd(&hist[key_of(tt) >> 16], 1u);
  }

  // Deterministic block reduction: WMMA wave sums, serial 8-way combine.
  __shared__ float shm[MAIN_TPB / 32];
  float vals[3] = {t1, t2, sd};
#pragma unroll
  for (int q = 0; q < 3; ++q) {
    float s = wave_sum32(vals[q]);
    if ((tid & 31) == 0) shm[tid >> 5] = s;
    __syncthreads();
    if (tid == 0) {
      float r = 0.0f;
      for (int w = 0; w < MAIN_TPB / 32; ++w) r += shm[w];
      partials[blockIdx.x * 3 + q] = r;
    }
    __syncthreads();
  }
}

// ---- pass 2: single-workgroup scan of the 65536-bin histogram ---------------
__global__ void __launch_bounds__(1024)
pass_scan(const unsigned int* __restrict__ hist,
          unsigned int* __restrict__ off,
          unsigned int* __restrict__ base,
          unsigned int* __restrict__ cursor, int n)
{
  __shared__ unsigned int seg[1024];
  const int t  = threadIdx.x;
  const int b0 = t * (BINS / 1024);          // 64 bins per thread
  unsigned int s = 0;
  for (int j = 0; j < BINS / 1024; ++j) s += hist[b0 + j];
  seg[t] = s;
  __syncthreads();
  if (t == 0) {
    unsigned int run = 0;
    for (int i = 0; i < 1024; ++i) { unsigned int v = seg[i]; seg[i] = run; run += v; }
  }
  __syncthreads();
  unsigned int run = seg[t];
  for (int j = 0; j < BINS / 1024; ++j) {
    const int b = b0 + j;
    const unsigned int h = hist[b];
    off[b]    = run;
    cursor[b] = run;
    base[b]   = (unsigned int)n - run - h;   // elements in strictly-greater buckets
    run += h;
  }
}

// ---- pass 3: scatter keys into bucket-grouped order -------------------------
__global__ void __launch_bounds__(SCAT_TPB)
pass_scatter(const float* __restrict__ targ, unsigned int* __restrict__ cursor,
             unsigned int* __restrict__ keys, int n)
{
  const int gid    = blockIdx.x * SCAT_TPB + threadIdx.x;
  const int stride = gridDim.x * SCAT_TPB;
  for (int i = gid; i < n; i += stride) {
    const uint32_t key = key_of(targ[i]);
    const unsigned int slot = atomicAdd(&cursor[key >> 16], 1u);
    keys[slot] = key;
  }
}

// ---- pass 4: one workgroup per bucket, exact ranks via tiled O(m^2) ---------
__global__ void __launch_bounds__(BUCKET_TPB)
pass_bucket(const unsigned int* __restrict__ hist,
            const unsigned int* __restrict__ off,
            const unsigned int* __restrict__ base,
            const unsigned int* __restrict__ keys,
            float* __restrict__ idcg_part)
{
  const int b   = blockIdx.x;
  const int tid = threadIdx.x;
  const unsigned int m = hist[b];
  if (m == 0) { if (tid == 0) idcg_part[b] = 0.0f; return; }
  const unsigned int start = off[b];
  const unsigned int brank = base[b];

  __shared__ unsigned int skey[TILE];
  double acc = 0.0;

  for (unsigned int jo = 0; jo < m; jo += BUCKET_TPB) {
    const unsigned int j = jo + (unsigned int)tid;
    const bool valid = (j < m);
    const unsigned int mykey = valid ? keys[start + j] : 0u;
    unsigned int c = 0;
    for (unsigned int to = 0; to < m; to += TILE) {
      unsigned int tc = m - to; if (tc > TILE) tc = TILE;
      __syncthreads();                                   // protect prior tile
      // Async DMA of this tile into LDS (one dword per active lane), then
      // drain ASYNCcnt before the workgroup barrier.
      if ((unsigned int)tid < tc) {
        async_copy_dword_to_lds(keys + start + to + tid, &skey[tid]);
      }
      wait_asynccnt0();
      __syncthreads();
      if (valid) {
        for (unsigned int k = 0; k < tc; ++k) {
          const unsigned int kk = skey[k];
          c += (kk > mykey) || (kk == mykey && (to + k) < j);  // ties: any stable order
        }
      }
    }
    if (valid) {
      const float rank1 = (float)(brank + c + 1u);             // 1-based global rank
      const float tf    = (float)(mykey >> 8) * (1.0f / 16777216.0f);  // exact
      const float g     = exp2f(tf) - 1.0f;
      acc += (double)(g * log2f(rank1 + 1.0f));
    }
  }

  // double accumulation -> f32 output is stable vs. scatter ordering
  for (int o = 16; o > 0; o >>= 1) acc += __shfl_xor(acc, o, 32);
  __shared__ double wred[BUCKET_TPB / 32];
  if ((tid & 31) == 0) wred[tid >> 5] = acc;
  __syncthreads();
  if (tid == 0) {
    double s = 0.0;
    for (int w = 0; w < BUCKET_TPB / 32; ++w) s += wred[w];
    idcg_part[b] = (float)s;
  }
}

// ---- pass 5: deterministic final combine ------------------------------------
__global__ void __launch_bounds__(FINAL_TPB)
pass_final(const float* __restrict__ idcg_part, const float* __restrict__ partials,
           float* __restrict__ out)
{
  const int tid = threadIdx.x;
  double di = 0.0, d1 = 0.0, d2 = 0.0, ds = 0.0;
  for (int i = tid; i < BINS; i += FINAL_TPB) di += (double)idcg_part[i];
  for (int w = tid; w < MAIN_WGS; w += FINAL_TPB) {
    d1 += (double)partials[w * 3 + 0];
    d2 += (double)partials[w * 3 + 1];
    ds += (double)partials[w * 3 + 2];
  }
  for (int o = 16; o > 0; o >>= 1) {
    di += __shfl_xor(di, o, 32);
    d1 += __shfl_xor(d1, o, 32);
    d2 += __shfl_xor(d2, o, 32);
    ds += __shfl_xor(ds, o, 32);
  }
  __shared__ double red[4][FINAL_TPB / 32];
  if ((tid & 31) == 0) {
    const int w = tid >> 5;
    red[0][w] = di; red[1][w] = d1; red[2][w] = d2; red[3][w] = ds;
  }
  __syncthreads();
  if (tid == 0) {
    double I = 0, T1 = 0, T2 = 0, S = 0;
    for (int w = 0; w < FINAL_TPB / 32; ++w) {
      I += red[0][w]; T1 += red[1][w]; T2 += red[2][w]; S += red[3][w];
    }
    const double approx_dcg = S * T1 + T2;
    out[0] = (float)(1.0 - approx_dcg / (I + 1e-10));
  }
}

// ---- host launcher ----------------------------------------------------------
extern "C" void kernel_launch(void* const* d_in, const int* in_sizes, int n_in,
                              void* d_out, int out_size, void* d_ws, size_t ws_size,
                              hipStream_t stream) {
  const float* pred = (const float*)d_in[0];
  const float* targ = (const float*)d_in[1];
  const int n = in_sizes[0];

  uint8_t* ws = (uint8_t*)d_ws;
  unsigned int* hist   = (unsigned int*)(ws);
  unsigned int* off    = (unsigned int*)(ws + 262144);
  unsigned int* base   = (unsigned int*)(ws + 524288);
  unsigned int* cursor = (unsigned int*)(ws + 786432);
  float*        idcgp  = (float*)(ws + 1048576);
  float*        parts  = (float*)(ws + 1310720);
  unsigned int* keys   = (unsigned int*)(ws + 1335296);  // 4*n bytes

  (void)hipMemsetAsync(hist, 0, BINS * sizeof(unsigned int), stream);  // capture-safe
  pass_main<<<MAIN_WGS, MAIN_TPB, 0, stream>>>(pred, targ, hist, parts, n);
  pass_scan<<<1, 1024, 0, stream>>>(hist, off, base, cursor, n);
  pass_scatter<<<SCAT_WGS, SCAT_TPB, 0, stream>>>(targ, cursor, keys, n);
  pass_bucket<<<BINS, BUCKET_TPB, 0, stream>>>(hist, off, base, keys, idcgp);
  pass_final<<<1, FINAL_TPB, 0, stream>>>(idcgp, parts, (float*)d_out);
}